// TotalProbabilityCell_58085137711593
// MI455X (gfx1250) — compile-verified
//
#include <hip/hip_runtime.h>
#include <stdint.h>

#define QS 64
#define DEPTH 16                // LDS tile ring depth (16KB each) -> 256KB in flight
#define NTHREADS 512            // 16 waves (wave32)

typedef unsigned int u32;
typedef u32 v4u __attribute__((ext_vector_type(4)));
typedef int  v4i __attribute__((ext_vector_type(4)));
typedef int  v8i __attribute__((ext_vector_type(8)));

__global__ __launch_bounds__(NTHREADS)
void hmm_forward_tdm_kernel(const float* __restrict__ cond,   // (B,T,64,64) log-space
                            const float* __restrict__ init,   // (B,64) log-space
                            float* __restrict__ out_seq,      // (B,T,64)
                            float* __restrict__ out_ll,       // (B,)
                            int T)
{
    __shared__ __align__(16) float tile[DEPTH][QS * QS];  // DEPTH x 16KB ring
    __shared__ float pm[8 * QS];                          // partial maxes
    __shared__ float ps[8 * QS];                          // partial sums
    __shared__ float fwd2[QS];                            // forward state, log2 space

    const int b   = blockIdx.x;
    const int tid = threadIdx.x;
    const int w   = tid >> 5;
    const int l   = tid & 31;
    const int j   = ((w & 1) << 5) | l;   // output state column (0..63)
    const int ig  = w >> 1;               // i-group 0..7 (8 rows each)
    const int ibase = ig << 3;

    // Scalar (SGPR) wave id -> guaranteed s_cbranch guards around TDM ops,
    // which ignore EXEC and must be reached by wave 0 only.
    const int ws = __builtin_amdgcn_readfirstlane(w);

    const float LOG2E = 1.4426950408889634f;
    const float LN2   = 0.6931471805599453f;

    if (tid < QS) fwd2[tid] = init[b * QS + tid] * LOG2E;

    // LDS byte offset of ring base (flat-address low 32 bits = LDS offset)
    const uint32_t lds_base0 = (uint32_t)(uintptr_t)(&tile[0][0]);

    const uint64_t gbase =
        (uint64_t)(uintptr_t)cond + (uint64_t)b * (uint64_t)T * (QS * QS * 4);
    const int tmax = T - 1;

    // D# group1 (constant): 4-byte elements, 1-row tile of 4096, stride 4096.
    //  w0: workgroup_mask=0, data_size=2 (4B)
    //  w1[31:16] = tensor_dim0.lo16 = 0x1000      w2[31:16] = tensor_dim1 = 1
    //  w3[31:16] = tile_dim0 = 0x1000             w4[15:0]  = tile_dim1 = 1
    //  w5 = tensor_dim0_stride = 4096             w6[31:16] = tensor_dim1_stride.lo16
    const v8i g1 = { 0x00020000, 0x10000000, 0x00010000, 0x10000000,
                     0x00000001, 0x00001000, 0x10000000, 0x00000000 };
    const v4i z4 = { 0, 0, 0, 0 };
    const v8i z8 = { 0, 0, 0, 0, 0, 0, 0, 0 };

    // Issue one 16KB tile DMA (wave-0 only; TENSORcnt is per-wave).
    auto issue = [&](int t) {
        uint64_t ga = gbase + (uint64_t)t * (QS * QS * 4);
        v4u g0;
        g0.x = 1u;                                           // count=1, no gather
        g0.y = lds_base0 + (u32)(t % DEPTH) * (QS * QS * 4); // lds_addr
        g0.z = (u32)ga;                                      // global_addr[31:0]
        g0.w = (u32)(ga >> 32) | (2u << 30);                 // global_addr[56:32] | type=2
        __builtin_amdgcn_tensor_load_to_lds(g0, g1, z4, z4, z8, 0);
    };

    if (ws == 0) {
#pragma unroll
        for (int d = 0; d < DEPTH; ++d) issue(d <= tmax ? d : tmax);
        __builtin_amdgcn_s_wait_tensorcnt(DEPTH - 1);   // buffer 0 resident
    }
    __syncthreads();  // fwd2 + buffer 0 visible to all waves

    for (int t = 0; t < T; ++t) {
        const float* tl = &tile[t % DEPTH][0];

        // x[k] = fwd2[i] + cond[i][j]*log2(e)   (log2 space), i = ibase+k
        float x[8];
#pragma unroll
        for (int k = 0; k < 8; ++k) {
            float c = tl[(ibase + k) * QS + j];          // lane-consecutive j: conflict-free
            x[k] = __builtin_fmaf(c, LOG2E, fwd2[ibase + k]);
        }
        float m = x[0];
#pragma unroll
        for (int k = 1; k < 8; ++k) m = fmaxf(m, x[k]);
        float s = 0.0f;
#pragma unroll
        for (int k = 0; k < 8; ++k) s += __builtin_amdgcn_exp2f(x[k] - m);

        pm[ig * QS + j] = m;
        ps[ig * QS + j] = s;
        __syncthreads();       // partials ready; all reads of buffer t%DEPTH done

        // Refill the just-consumed buffer (clamp near the tail keeps exactly
        // DEPTH descriptors outstanding, so the wait threshold is constant).
        if (ws == 0) {
            int tn = t + DEPTH;
            issue(tn <= tmax ? tn : tmax);
        }

        if (tid < QS) {        // combine 8 partials per column, emit forward_seq
            float mm = pm[tid];
#pragma unroll
            for (int g = 1; g < 8; ++g) mm = fmaxf(mm, pm[g * QS + tid]);
            float ss = 0.0f;
#pragma unroll
            for (int g = 0; g < 8; ++g)
                ss += ps[g * QS + tid] * __builtin_amdgcn_exp2f(pm[g * QS + tid] - mm);
            float r2 = mm + __builtin_amdgcn_logf(ss);   // v_log_f32 = log2
            fwd2[tid] = r2;
            out_seq[((size_t)b * T + t) * QS + tid] = r2 * LN2;
        }

        // Retire buffer (t+1)%DEPTH now so one barrier publishes both the new
        // fwd2 and the next tile (2 barriers/step instead of 3).
        if (ws == 0) __builtin_amdgcn_s_wait_tensorcnt(DEPTH - 1);
        __syncthreads();
    }

    if (tid == 0) {            // loglik = logsumexp over final state
        float mm = fwd2[0];
        for (int q = 1; q < QS; ++q) mm = fmaxf(mm, fwd2[q]);
        float ss = 0.0f;
        for (int q = 0; q < QS; ++q) ss += __builtin_amdgcn_exp2f(fwd2[q] - mm);
        out_ll[b] = (mm + __builtin_amdgcn_logf(ss)) * LN2;
    }
}

extern "C" void kernel_launch(void* const* d_in, const int* in_sizes, int n_in,
                              void* d_out, int out_size, void* d_ws, size_t ws_size,
                              hipStream_t stream) {
    (void)n_in; (void)d_ws; (void)ws_size; (void)out_size;
    const float* cond = (const float*)d_in[0];   // (B,T,Q*Q) f32
    const float* init = (const float*)d_in[1];   // (B,Q) f32
    const int B = in_sizes[1] / QS;
    const int T = in_sizes[0] / (B * QS * QS);
    float* out_seq = (float*)d_out;
    float* out_ll  = out_seq + (size_t)B * T * QS;
    hipLaunchKernelGGL(hmm_forward_tdm_kernel, dim3(B), dim3(NTHREADS), 0, stream,
                       cond, init, out_seq, out_ll, T);
}